// LDPC5GDecoder_60653528154574
// MI455X (gfx1250) — compile-verified
//
#include <hip/hip_runtime.h>
#include <cstdint>
#include <cstddef>

// ---------------- Problem constants (fixed by the reference) ----------------
#define Zl     384                 // lifting factor
#define MBASE  46                  // basegraph rows
#define NBASE  68                  // basegraph cols
#define DEG    7                   // check-node degree
#define KINFO  8448                // info bits
#define NCH    25344               // transmitted bits per batch element
#define NLDPC  (NBASE * Zl)        // 26112 variable nodes
#define MCHK   (MBASE * Zl)        // 17664 check nodes
#define NBLK   (MBASE * DEG)       // 322 lifted edge-blocks
#define NEDGE  (NBLK * Zl)         // 123648 edges
#define BSZ    128                 // batch
#define NITER  20
#define LLRMAX 20.0f
#define MAXVDEG 32                 // safety bound on VN base-degree (actual ~<=12)
#define STAGE_MAX 8                // rows staged in LDS via async copy per node

// ---- gfx1250 async-to-LDS path (builtin confirmed present by round-2 probe) ----
#if defined(__has_builtin)
#if __has_builtin(__builtin_amdgcn_global_load_async_to_lds_b128)
#define HAVE_ASYNC 1
#endif
#endif
#ifndef HAVE_ASYNC
#define HAVE_ASYNC 0
#endif

#if HAVE_ASYNC
typedef int v4i __attribute__((ext_vector_type(4)));
typedef __attribute__((address_space(3))) v4i* lds_v4i_ptr;

__device__ __forceinline__ void async_copy_b128(const float* g, float* l) {
  // per-lane: copies 16 bytes from global address g to LDS address l; tracked by ASYNCcnt
  __builtin_amdgcn_global_load_async_to_lds_b128(
      (v4i*)g,            // global source (param type: v4i*, non-const, per diag)
      (lds_v4i_ptr)l,     // LDS dest (AS3; implicitly converts if param is generic)
      0, 0);              // imm offset, imm cpol
}
#if __has_builtin(__builtin_amdgcn_s_wait_asynccnt)
#define WAIT_ASYNC() __builtin_amdgcn_s_wait_asynccnt(0)
#else
#define WAIT_ASYNC() asm volatile("s_wait_asynccnt 0x0" ::: "memory")
#endif
#endif

// ---------------------------------------------------------------------------
// Build per-base-column adjacency from col[]:  edge block b (b = r*DEG + j)
// satisfies col[b*Z + i] = c*Z + (i + shift) % Z, so col[b*Z] = c*Z + shift.
// Serial single thread => deterministic list order (ascending b == ascending
// edge id, matching the reference's segment order).
// ---------------------------------------------------------------------------
__global__ void ldpc_build_meta(const int* __restrict__ col,
                                int* __restrict__ counts,
                                int* __restrict__ entries) {
  if (blockIdx.x == 0 && threadIdx.x == 0) {
    for (int c = 0; c < NBASE; ++c) counts[c] = 0;
    for (int b = 0; b < NBLK; ++b) {
      int cv = col[b * Zl];
      int c  = cv / Zl;
      int sh = cv - c * Zl;
      int d  = counts[c];
      if (d < MAXVDEG) {
        entries[c * MAXVDEG + d] = (b << 16) | sh;
        counts[c] = d + 1;
      }
    }
  }
}

// ---------------------------------------------------------------------------
// Init: lch[v][b] = -clip(llr_ch[b][v-2Z]); punctured (v < 2Z) -> 0.
// Also seeds vn_total = lch (== first VN update with m_cv = 0).
// ---------------------------------------------------------------------------
__global__ __launch_bounds__(256)
void ldpc_init(const float* __restrict__ llr_ch,
               float* __restrict__ lch,
               float* __restrict__ vn) {
  int tid  = blockIdx.x * blockDim.x + threadIdx.x;
  int v    = tid >> 5;
  int lane = tid & 31;
  if (v >= NLDPC) return;
  size_t base = (size_t)v * BSZ + lane * 4;

  float4 r = make_float4(0.f, 0.f, 0.f, 0.f);
  if (v >= 2 * Zl) {
    int n  = v - 2 * Zl;
    int b0 = lane * 4;
    float a0 = llr_ch[(size_t)(b0 + 0) * NCH + n];
    float a1 = llr_ch[(size_t)(b0 + 1) * NCH + n];
    float a2 = llr_ch[(size_t)(b0 + 2) * NCH + n];
    float a3 = llr_ch[(size_t)(b0 + 3) * NCH + n];
    r.x = -fminf(fmaxf(a0, -LLRMAX), LLRMAX);
    r.y = -fminf(fmaxf(a1, -LLRMAX), LLRMAX);
    r.z = -fminf(fmaxf(a2, -LLRMAX), LLRMAX);
    r.w = -fminf(fmaxf(a3, -LLRMAX), LLRMAX);
  }
  *(float4*)(lch + base) = r;
  *(float4*)(vn  + base) = r;
}

// ---------------------------------------------------------------------------
// Check-node update (min-sum, exact reference tie rule: first argmin gets
// min2, others min1; sign = parity excluding self, neg strictly x < 0).
// Wave32 per check node; lane = float4 of batch. first => old m_cv == 0.
// Direct 512B-coalesced loads (data consumed once -> no LDS staging here).
// ---------------------------------------------------------------------------
__global__ __launch_bounds__(256)
void ldpc_cn(const float* __restrict__ vn,
             float* __restrict__ mcv,
             const int* __restrict__ col,
             int first) {
  int tid  = blockIdx.x * blockDim.x + threadIdx.x;
  int m    = tid >> 5;
  int lane = tid & 31;
  if (m >= MCHK) return;
  int i    = m % Zl;
  int r    = m / Zl;
  int boff = lane * 4;

  int e[DEG];
#pragma unroll
  for (int j = 0; j < DEG; ++j) e[j] = (r * DEG + j) * Zl + i;

  float tt[DEG][4];
#pragma unroll
  for (int j = 0; j < DEG; ++j) {
    int vv  = col[e[j]];                       // wave-uniform
    float4 q = *(const float4*)(vn + (size_t)vv * BSZ + boff);
    tt[j][0] = q.x; tt[j][1] = q.y; tt[j][2] = q.z; tt[j][3] = q.w;
  }

  float cc[DEG][4];
  if (!first) {
#pragma unroll
    for (int j = 0; j < DEG; ++j) {
      float4 q = *(const float4*)(mcv + (size_t)e[j] * BSZ + boff);
      cc[j][0] = q.x; cc[j][1] = q.y; cc[j][2] = q.z; cc[j][3] = q.w;
    }
  } else {
#pragma unroll
    for (int j = 0; j < DEG; ++j) {
      cc[j][0] = 0.f; cc[j][1] = 0.f; cc[j][2] = 0.f; cc[j][3] = 0.f;
    }
  }

#pragma unroll
  for (int comp = 0; comp < 4; ++comp) {
    float mv[DEG];
    float min1 = 1e30f, min2 = 1e30f;
    int amin = 0, par = 0;
#pragma unroll
    for (int j = 0; j < DEG; ++j) {
      float x = tt[j][comp] - cc[j][comp];     // m_vc (extrinsic VN message)
      mv[j] = x;
      float a = fabsf(x);
      if (x < 0.0f) par ^= 1;
      if (a < min1)      { min2 = min1; min1 = a; amin = j; }
      else if (a < min2) { min2 = a; }
    }
#pragma unroll
    for (int j = 0; j < DEG; ++j) {
      float a   = (j == amin) ? min2 : min1;   // exclude-self min
      int   neg = (mv[j] < 0.0f) ? 1 : 0;
      cc[j][comp] = ((par ^ neg) != 0) ? -a : a;
    }
  }

#pragma unroll
  for (int j = 0; j < DEG; ++j) {
    float4 q;
    q.x = cc[j][0]; q.y = cc[j][1]; q.z = cc[j][2]; q.w = cc[j][3];
    *(float4*)(mcv + (size_t)e[j] * BSZ + boff) = q;
  }
}

// ---------------------------------------------------------------------------
// Variable-node update: vn[v] = lch[v] + sum of incident m_cv in ascending-
// edge order (deterministic). Wave32 per variable node.
// gfx1250 path: incident message rows are copied global->LDS with
// GLOBAL_LOAD_ASYNC_TO_LDS_B128 (ASYNCcnt), overlapped with the lch load,
// then consumed from LDS after s_wait_asynccnt 0. Wave-private staging, so
// no workgroup barrier is needed and values are bit-identical.
// ---------------------------------------------------------------------------
__global__ __launch_bounds__(256)
void ldpc_vn(const float* __restrict__ lch,
             const float* __restrict__ mcv,
             float* __restrict__ vn,
             const int* __restrict__ counts,
             const int* __restrict__ entries) {
  __shared__ int s_cnt[NBASE];
  __shared__ int s_ent[NBASE * MAXVDEG];
#if HAVE_ASYNC
  __shared__ float s_stage[8][STAGE_MAX][BSZ];   // 32 KB, per-wave-private rows
#endif
  for (int x = threadIdx.x; x < NBASE; x += blockDim.x) s_cnt[x] = counts[x];
  for (int x = threadIdx.x; x < NBASE * MAXVDEG; x += blockDim.x) s_ent[x] = entries[x];
  __syncthreads();

  int tid  = blockIdx.x * blockDim.x + threadIdx.x;
  int v    = tid >> 5;
  int lane = tid & 31;
  if (v >= NLDPC) return;
  int boff  = lane * 4;
  int cb    = v / Zl;
  int k     = v - cb * Zl;
  int deg   = s_cnt[cb];
  int mbase = cb * MAXVDEG;

#if HAVE_ASYNC
  int w = threadIdx.x >> 5;
  int nst = (deg < STAGE_MAX) ? deg : STAGE_MAX;
  for (int d = 0; d < nst; ++d) {
    int p   = s_ent[mbase + d];
    int idx = k - (p & 0xFFFF);
    idx += (idx < 0) ? Zl : 0;
    int e   = (p >> 16) * Zl + idx;
    async_copy_b128(mcv + (size_t)e * BSZ + boff, &s_stage[w][d][boff]);
  }
#else
  // fallback: prefetch pass (global_prefetch_b8)
  for (int d = 0; d < deg; ++d) {
    int p   = s_ent[mbase + d];
    int idx = k - (p & 0xFFFF);
    idx += (idx < 0) ? Zl : 0;
    int e   = (p >> 16) * Zl + idx;
    __builtin_prefetch(mcv + (size_t)e * BSZ + boff, 0, 0);
  }
#endif

  float4 acc = *(const float4*)(lch + (size_t)v * BSZ + boff);

#if HAVE_ASYNC
  WAIT_ASYNC();
#endif

  for (int d = 0; d < deg; ++d) {
    int p   = s_ent[mbase + d];
    int idx = k - (p & 0xFFFF);
    idx += (idx < 0) ? Zl : 0;
    int e   = (p >> 16) * Zl + idx;
    float4 q;
#if HAVE_ASYNC
    if (d < nst) q = *(const float4*)&s_stage[w][d][boff];
    else         q = *(const float4*)(mcv + (size_t)e * BSZ + boff);
#else
    q = *(const float4*)(mcv + (size_t)e * BSZ + boff);
#endif
    acc.x += q.x; acc.y += q.y; acc.z += q.z; acc.w += q.w;
  }
  *(float4*)(vn + (size_t)v * BSZ + boff) = acc;
}

// ---------------------------------------------------------------------------
// Output: llr_out[b][v] = -x_tot[v][b] for v < K.  Coalesced stores.
// ---------------------------------------------------------------------------
__global__ __launch_bounds__(256)
void ldpc_out(const float* __restrict__ vn, float* __restrict__ out) {
  int tid = blockIdx.x * blockDim.x + threadIdx.x;
  if (tid >= KINFO * BSZ) return;
  int b = tid / KINFO;
  int v = tid - b * KINFO;
  out[tid] = -vn[(size_t)v * BSZ + b];
}

// ---------------------------------------------------------------------------
extern "C" void kernel_launch(void* const* d_in, const int* in_sizes, int n_in,
                              void* d_out, int out_size, void* d_ws, size_t ws_size,
                              hipStream_t stream) {
  (void)in_sizes; (void)n_in; (void)out_size; (void)ws_size;
  const float* llr_ch = (const float*)d_in[0];
  const int*   col    = (const int*)d_in[2];   // d_in[1] = row (implied by QC structure)

  // workspace layout (~90.1 MB; L2-resident on MI455X's 192 MB global L2)
  float* mcv     = (float*)d_ws;                          // NEDGE * 128
  float* vn      = mcv + (size_t)NEDGE * BSZ;             // NLDPC * 128
  float* lch     = vn  + (size_t)NLDPC * BSZ;             // NLDPC * 128
  int*   counts  = (int*)(lch + (size_t)NLDPC * BSZ);     // NBASE
  int*   entries = counts + NBASE;                        // NBASE * MAXVDEG

  ldpc_build_meta<<<1, 1, 0, stream>>>(col, counts, entries);
  ldpc_init<<<(NLDPC * 32) / 256, 256, 0, stream>>>(llr_ch, lch, vn);

  for (int it = 0; it < NITER; ++it) {
    ldpc_cn<<<(MCHK * 32) / 256, 256, 0, stream>>>(vn, mcv, col, (it == 0) ? 1 : 0);
    ldpc_vn<<<(NLDPC * 32) / 256, 256, 0, stream>>>(lch, mcv, vn, counts, entries);
  }

  ldpc_out<<<(KINFO * BSZ) / 256, 256, 0, stream>>>(vn, (float*)d_out);
}